// Compositional_86028194939366
// MI455X (gfx1250) — compile-verified
//
#include <hip/hip_runtime.h>
#include <cmath>

// ---------------------------------------------------------------------------
// Banded GEMM + bias + tanh for MI455X (gfx1250), wave32 WMMA.
//
//   y[b, j] = tanh( sum_r x[b, r] * W[r, j] * band(r, j) + bias[j] )
//   band(r, j) nonzero iff  j/u <= r <= j/u + g
//
// For a 16-wide column tile starting at j0, the nonzero K-rows are
// r0 .. r0 + (16/u + g - 1), r0 = j0/u (12 rows for u=4, g=8).  We run
// KSTEP_MAX=4 chained V_WMMA_F32_16X16X4_F32 steps (16-row K window) with
// out-of-band B rows zeroed via select, so the hot loop is branch-free.
// Masked W fragments load once per wave and are reused across 16 batch
// tiles.  Kernel is HBM-store bound (~256 MB out + ~64 MB x vs ~1.2 GFLOP,
// roofline ~15 us @ 23.3 TB/s); epilogue uses gfx1250 hardware v_tanh_f32.
// Loads/stores go through address_space(1) pointers so the backend emits
// global_load/global_store (LOADcnt only) instead of flat ops that also
// tick DScnt and occupy the LDS issue path.  The prefetch uses a generic
// pointer because __builtin_prefetch rejects AS(1) args in device code.
// ---------------------------------------------------------------------------

typedef __attribute__((ext_vector_type(2))) float v2f;
typedef __attribute__((ext_vector_type(8))) float v8f;

#define GLOBAL_AS __attribute__((address_space(1)))
typedef const GLOBAL_AS float* gcfp;
typedef       GLOBAL_AS float* gfp;
typedef const GLOBAL_AS v2f*   gcv2p;
typedef const GLOBAL_AS int*   gcip;

#define KSTEP_MAX 4   // K window = 16 rows: supports 16/u + g <= 16

__device__ __forceinline__ float fast_tanh(float y) {
#if __has_builtin(__builtin_amdgcn_tanhf)
  return __builtin_amdgcn_tanhf(y);          // single v_tanh_f32
#else
  float ay = __builtin_fabsf(y);
  float e  = __expf(-2.0f * ay);
  float r  = __fdividef(1.0f - e, 1.0f + e);
  return __builtin_copysignf(r, y);
#endif
}

// ---------------- fully specialized fast path (compile-time shapes) --------
template <int D, int N, int U, int TPW>
__global__ __launch_bounds__(256) void band_gemm_tanh_wmma_tpl(
    const float* __restrict__ x_,     // [B, D]
    const float* __restrict__ W_,     // [D, N]
    const float* __restrict__ bias_,  // [N]
    const int*   __restrict__ gp_,    // graph_distance (device scalar)
    float*       __restrict__ out_)   // [B, N]
{
  gcfp x    = (gcfp)x_;
  gcfp W    = (gcfp)W_;
  gcfp bias = (gcfp)bias_;
  gfp  out  = (gfp)out_;

  const int g    = ((gcip)gp_)[0];
  const int lane = threadIdx.x & 31;
  const int wave = threadIdx.x >> 5;
  const int n    = lane & 15;
  const int h    = lane >> 4;

  const int j0   = blockIdx.x * 16;
  const int r0   = j0 / U;
  const int cdiv = n / U;

  // Masked W (B-matrix) fragments: loaded once, reused for all batch tiles.
  v2f Bf[KSTEP_MAX];
#pragma unroll
  for (int t = 0; t < KSTEP_MAX; ++t) {
    const int k0 = 4 * t + 2 * h;
    v2f bf;
#pragma unroll
    for (int q = 0; q < 2; ++q) {
      const int k = k0 + q;
      int row = r0 + k; if (row > D - 1) row = D - 1;         // safe address
      const float w = W[(size_t)row * N + (j0 + n)];
      const bool inband = (k >= cdiv) && (k <= cdiv + g) && (r0 + k < D);
      const float v = inband ? w : 0.0f;
      if (q == 0) bf.x = v; else bf.y = v;
    }
    Bf[t] = bf;
  }

  const float bv = bias[j0 + n];

  const int waveLinear = blockIdx.y * 8 + wave;   // 8 waves per 256-thread block
  const int btBase     = waveLinear * TPW;

  // Per-fragment x pointers (clamped to keep b64 loads aligned & in-bounds;
  // clamped rows only pair with zero B rows, and x is all-finite).
  gcfp xrow = x + (size_t)(btBase * 16 + n) * D;
  gcfp xp[KSTEP_MAX];
  int idx0 = 0;
#pragma unroll
  for (int t = 0; t < KSTEP_MAX; ++t) {
    int idx = r0 + 4 * t + 2 * h;
    if (idx > D - 2) idx = D - 2;
    if (t == 0) idx0 = idx;
    xp[t] = xrow + idx;
  }
  // Generic-pointer mirror of xp[0], used only for prefetch.
  const float* pf = x_ + (size_t)(btBase * 16 + n) * D + idx0;

  gfp op = out + (size_t)(btBase * 16 + 8 * h) * N + (size_t)(j0 + n);

#pragma unroll 1
  for (int i = 0; i < TPW; ++i) {
    // Speculative prefetch of next tile's window (overshoot is dropped).
    __builtin_prefetch(pf + (size_t)16 * D, 0, 0);

    v2f Af[KSTEP_MAX];
#pragma unroll
    for (int t = 0; t < KSTEP_MAX; ++t) Af[t] = *(gcv2p)xp[t];

    v8f acc = {0.f, 0.f, 0.f, 0.f, 0.f, 0.f, 0.f, 0.f};
#pragma unroll
    for (int t = 0; t < KSTEP_MAX; ++t) {
      acc = __builtin_amdgcn_wmma_f32_16x16x4_f32(
          false, Af[t], false, Bf[t], (short)0, acc, false, false);
    }

    // C/D layout: VGPR v, half h -> row v + 8h, col n.  N is compile-time,
    // so these 8 stores use instruction immediate offsets off one base.
#pragma unroll
    for (int v = 0; v < 8; ++v) {
      op[(size_t)v * N] = fast_tanh(acc[v] + bv);
    }

#pragma unroll
    for (int t = 0; t < KSTEP_MAX; ++t) xp[t] += (size_t)16 * D;
    pf += (size_t)16 * D;
    op += (size_t)16 * N;
  }
}

// ---------------- runtime-generic fallback ---------------------------------
__global__ __launch_bounds__(256) void band_gemm_tanh_wmma_gen(
    const float* __restrict__ x_, const float* __restrict__ W_,
    const float* __restrict__ bias_, const int* __restrict__ gp_,
    float* __restrict__ out_, int B, int D, int N, int u, int tilesPerWave)
{
  gcfp x    = (gcfp)x_;
  gcfp W    = (gcfp)W_;
  gcfp bias = (gcfp)bias_;
  gfp  out  = (gfp)out_;

  const int g    = ((gcip)gp_)[0];
  const int lane = threadIdx.x & 31;
  const int wave = threadIdx.x >> 5;
  const int n    = lane & 15;
  const int h    = lane >> 4;

  const int j0   = blockIdx.x * 16;
  const int r0   = j0 / u;
  const int cdiv = n / u;

  v2f Bf[KSTEP_MAX];
#pragma unroll
  for (int t = 0; t < KSTEP_MAX; ++t) {
    const int k0 = 4 * t + 2 * h;
    v2f bf;
#pragma unroll
    for (int q = 0; q < 2; ++q) {
      const int k = k0 + q;
      int row = r0 + k; if (row > D - 1) row = D - 1;
      const float w = W[(size_t)row * N + (j0 + n)];
      const bool inband = (k >= cdiv) && (k <= cdiv + g) && (r0 + k < D);
      const float v = inband ? w : 0.0f;
      if (q == 0) bf.x = v; else bf.y = v;
    }
    Bf[t] = bf;
  }

  const float bv = bias[j0 + n];

  const int waveLinear  = blockIdx.y * (blockDim.x >> 5) + wave;
  const int nBatchTiles = B >> 4;
  const int btBase      = waveLinear * tilesPerWave;
  int cnt = nBatchTiles - btBase;
  if (cnt > tilesPerWave) cnt = tilesPerWave;
  if (cnt < 0) cnt = 0;

  gcfp xrow = x + (size_t)(btBase * 16 + n) * D;
  gcfp xp[KSTEP_MAX];
  int idx0 = 0;
#pragma unroll
  for (int t = 0; t < KSTEP_MAX; ++t) {
    int idx = r0 + 4 * t + 2 * h;
    if (idx > D - 2) idx = D - 2;
    if (t == 0) idx0 = idx;
    xp[t] = xrow + idx;
  }
  const float* pf = x_ + (size_t)(btBase * 16 + n) * D + idx0;

  gfp op = out + (size_t)(btBase * 16 + 8 * h) * N + (size_t)(j0 + n);

  for (int i = 0; i < cnt; ++i) {
    __builtin_prefetch(pf + (size_t)16 * D, 0, 0);

    v2f Af[KSTEP_MAX];
#pragma unroll
    for (int t = 0; t < KSTEP_MAX; ++t) Af[t] = *(gcv2p)xp[t];

    v8f acc = {0.f, 0.f, 0.f, 0.f, 0.f, 0.f, 0.f, 0.f};
#pragma unroll
    for (int t = 0; t < KSTEP_MAX; ++t) {
      acc = __builtin_amdgcn_wmma_f32_16x16x4_f32(
          false, Af[t], false, Bf[t], (short)0, acc, false, false);
    }

#pragma unroll
    for (int v = 0; v < 8; ++v) {
      op[(size_t)v * N] = fast_tanh(acc[v] + bv);
    }

#pragma unroll
    for (int t = 0; t < KSTEP_MAX; ++t) xp[t] += (size_t)16 * D;
    pf += (size_t)16 * D;
    op += (size_t)16 * N;
  }
}

extern "C" void kernel_launch(void* const* d_in, const int* in_sizes, int n_in,
                              void* d_out, int out_size, void* d_ws, size_t ws_size,
                              hipStream_t stream) {
  const float* x    = (const float*)d_in[0];
  const float* W    = (const float*)d_in[1];
  const float* bias = (const float*)d_in[2];
  const int* g_ptr  = (const int*)d_in[4];   // graph_distance (device scalar)

  const int N = in_sizes[2];           // D * u  (8192)
  const int D = in_sizes[1] / N;       // 2048
  const int B = in_sizes[0] / D;       // 8192
  const int u = N / D;                 // 4

  const int nColTiles   = N / 16;
  const int nBatchTiles = B / 16;
  const int WAVES_PER_BLOCK = 8;       // 256 threads, wave32
  const int TPW = 16;                  // batch tiles per wave

  if (D == 2048 && N == 8192 && u == 4 &&
      nBatchTiles % (TPW * WAVES_PER_BLOCK) == 0) {
    const int gy = nBatchTiles / (TPW * WAVES_PER_BLOCK);   // 4
    dim3 grid(nColTiles, gy, 1);
    band_gemm_tanh_wmma_tpl<2048, 8192, 4, 16>
        <<<grid, dim3(256, 1, 1), 0, stream>>>(x, W, bias, g_ptr, (float*)d_out);
  } else {
    const int wavesNeeded = (nBatchTiles + TPW - 1) / TPW;
    const int gy = (wavesNeeded + WAVES_PER_BLOCK - 1) / WAVES_PER_BLOCK;
    dim3 grid(nColTiles, gy, 1);
    band_gemm_tanh_wmma_gen<<<grid, dim3(256, 1, 1), 0, stream>>>(
        x, W, bias, g_ptr, (float*)d_out, B, D, N, u, TPW);
  }
}